// Classifier_10720238370855
// MI455X (gfx1250) — compile-verified
//
#include <hip/hip_runtime.h>
#include <math.h>

#define NNODES  40000
#define NEDGES  640000
#define DIM     128
#define NGRAPH  64
#define NCLS    100
#define NLAYERS 3

typedef __attribute__((ext_vector_type(2))) float v2f;
typedef __attribute__((ext_vector_type(8))) float v8f;

__device__ __forceinline__ void atomAddF(float* p, float v) {
    __hip_atomic_fetch_add(p, v, __ATOMIC_RELAXED, __HIP_MEMORY_SCOPE_AGENT);
}

// ---------------------------------------------------------------------------
// 1) nh[row] = h[row] / (||h[row]|| + 1e-12)   — one wave32 per row
// ---------------------------------------------------------------------------
__global__ void normalize_rows(const float* __restrict__ h,
                               float* __restrict__ nh, int n) {
    int lane = threadIdx.x & 31;
    int row  = blockIdx.x * (blockDim.x >> 5) + (threadIdx.x >> 5);
    if (row >= n) return;
    float4 v = ((const float4*)(h + (size_t)row * DIM))[lane];
    float ss = v.x * v.x + v.y * v.y + v.z * v.z + v.w * v.w;
#pragma unroll
    for (int off = 16; off >= 1; off >>= 1) ss += __shfl_xor(ss, off, 32);
    float inv = 1.0f / (sqrtf(ss) + 1e-12f);
    float4 o;
    o.x = v.x * inv; o.y = v.y * inv; o.z = v.z * inv; o.w = v.w * inv;
    ((float4*)(nh + (size_t)row * DIM))[lane] = o;
}

// ---------------------------------------------------------------------------
// 2) Fused edge pass — one wave32 per edge.
//    s = beta * dot(nh[src], nh[dst]);  a = exp(s)   (cosine bounded => no
//    max-subtraction needed; mathematically identical softmax)
//    denom[dst] += a;  num[dst][:] += a * h[src][:]  (L2-resident atomics)
// ---------------------------------------------------------------------------
__global__ void edge_attend(const float* __restrict__ nh,
                            const float* __restrict__ h,
                            const int* __restrict__ src,
                            const int* __restrict__ dst,
                            const float* __restrict__ betas, int layer,
                            float* __restrict__ num,
                            float* __restrict__ denom, int e) {
    int lane = threadIdx.x & 31;
    int edge = blockIdx.x * (blockDim.x >> 5) + (threadIdx.x >> 5);
    if (edge >= e) return;
    int s = src[edge];
    int d = dst[edge];
    float4 a4 = ((const float4*)(nh + (size_t)s * DIM))[lane];
    float4 b4 = ((const float4*)(nh + (size_t)d * DIM))[lane];
    float p = a4.x * b4.x + a4.y * b4.y + a4.z * b4.z + a4.w * b4.w;
#pragma unroll
    for (int off = 16; off >= 1; off >>= 1) p += __shfl_xor(p, off, 32);
    float att = __expf(betas[layer] * p);
    if (lane == 0) atomAddF(denom + d, att);
    float4 hv = ((const float4*)(h + (size_t)s * DIM))[lane];
    float* np = num + (size_t)d * DIM + lane * 4;
    atomAddF(np + 0, att * hv.x);
    atomAddF(np + 1, att * hv.y);
    atomAddF(np + 2, att * hv.z);
    atomAddF(np + 3, att * hv.w);
}

// ---------------------------------------------------------------------------
// 3) h_out = num / (denom > 0 ? denom : 1)     (in place on num)
// ---------------------------------------------------------------------------
__global__ void finalize_div(float* __restrict__ num,
                             const float* __restrict__ denom, int nd) {
    int i = blockIdx.x * blockDim.x + threadIdx.x;
    if (i >= nd) return;
    float dn = denom[i >> 7];
    num[i] = num[i] / (dn > 0.0f ? dn : 1.0f);
}

// ---------------------------------------------------------------------------
// 4) AvgPool readout
// ---------------------------------------------------------------------------
__global__ void pool_sum(const float* __restrict__ h,
                         const int* __restrict__ gid,
                         float* __restrict__ hg, int n) {
    int i = blockIdx.x * blockDim.x + threadIdx.x;
    if (i >= n * DIM) return;
    int node = i >> 7;
    int d    = i & 127;
    atomAddF(hg + (size_t)gid[node] * DIM + d, h[i]);
}

__global__ void pool_cnt(const int* __restrict__ gid,
                         float* __restrict__ cnt, int n) {
    int i = blockIdx.x * blockDim.x + threadIdx.x;
    if (i >= n) return;
    atomAddF(cnt + gid[i], 1.0f);
}

__global__ void pool_div(float* __restrict__ hg, const float* __restrict__ cnt) {
    int i = blockIdx.x * blockDim.x + threadIdx.x;
    if (i >= NGRAPH * DIM) return;
    hg[i] /= fmaxf(cnt[i >> 7], 1.0f);
}

// ---------------------------------------------------------------------------
// 5) Classifier: out[64x100] = hg[64x128] @ W[128x100] + b
//    v_wmma_f32_16x16x4_f32, one wave per 16x16 tile; N padded 100 -> 112
//    with masked (EXEC-preserving) operand loads.
//    A 16x4 f32 layout: lanes 0-15 -> M=lane, K=k0+{0,1}; lanes 16-31 ->
//    M=lane-16, K=k0+{2,3}.  B 4x16: rows striped the same way across halves.
//    D 16x16 f32: VGPR r -> rows r (lanes 0-15) / r+8 (lanes 16-31).
// ---------------------------------------------------------------------------
__global__ void classifier_wmma(const float* __restrict__ hg,
                                const float* __restrict__ W,
                                const float* __restrict__ b,
                                float* __restrict__ out) {
    int mt = blockIdx.x / 7;
    int nt = blockIdx.x % 7;
    int m0 = mt * 16;
    int n0 = nt * 16;
    int lane = threadIdx.x & 31;
    int half = lane >> 4;
    int lr   = lane & 15;
    int col  = n0 + lr;
    int ccol = col < NCLS ? col : (NCLS - 1);   // clamp address, mask value
    float mask = col < NCLS ? 1.0f : 0.0f;

    v8f acc = {};
#pragma unroll 4
    for (int k0 = 0; k0 < DIM; k0 += 4) {
        const float* ar = hg + (size_t)(m0 + lr) * DIM + k0 + 2 * half;
        v2f a;
        a.x = ar[0];
        a.y = ar[1];
        v2f bb;
        bb.x = W[(size_t)(k0 + 2 * half) * NCLS + ccol] * mask;
        bb.y = W[(size_t)(k0 + 2 * half + 1) * NCLS + ccol] * mask;
        acc = __builtin_amdgcn_wmma_f32_16x16x4_f32(
            /*neg_a=*/false, a, /*neg_b=*/false, bb,
            /*c_mod=*/(short)0, acc, /*reuse_a=*/false, /*reuse_b=*/false);
    }

    if (col < NCLS) {
        float bias = b[col];
#pragma unroll
        for (int r = 0; r < 8; ++r) {
            int m = m0 + r + 8 * half;
            out[(size_t)m * NCLS + col] = acc[r] + bias;
        }
    }
}

// ---------------------------------------------------------------------------
extern "C" void kernel_launch(void* const* d_in, const int* in_sizes, int n_in,
                              void* d_out, int out_size, void* d_ws, size_t ws_size,
                              hipStream_t stream) {
    const float* h0    = (const float*)d_in[0];
    const int*   src   = (const int*)d_in[1];
    const int*   dst   = (const int*)d_in[2];
    const int*   gid   = (const int*)d_in[3];
    const float* betas = (const float*)d_in[4];
    const float* Wc    = (const float*)d_in[5];
    const float* bc    = (const float*)d_in[6];
    float*       out   = (float*)d_out;

    // workspace layout
    float* nh    = (float*)d_ws;                         // N*D
    float* hA    = nh + (size_t)NNODES * DIM;            // N*D
    float* hB    = hA + (size_t)NNODES * DIM;            // N*D
    float* denom = hB + (size_t)NNODES * DIM;            // N
    float* hg    = denom + NNODES;                       // G*D
    float* cnt   = hg + NGRAPH * DIM;                    // G

    const int ND = NNODES * DIM;

    const float* hin = h0;
    float* outs[NLAYERS] = {hA, hB, hA};
    for (int l = 0; l < NLAYERS; ++l) {
        float* hout = outs[l];
        hipMemsetAsync(hout, 0, sizeof(float) * (size_t)ND, stream);
        hipMemsetAsync(denom, 0, sizeof(float) * NNODES, stream);
        normalize_rows<<<(NNODES + 7) / 8, 256, 0, stream>>>(hin, nh, NNODES);
        edge_attend<<<(NEDGES + 7) / 8, 256, 0, stream>>>(nh, hin, src, dst,
                                                          betas, l, hout, denom,
                                                          NEDGES);
        finalize_div<<<(ND + 255) / 256, 256, 0, stream>>>(hout, denom, ND);
        hin = hout;
    }

    hipMemsetAsync(hg, 0, sizeof(float) * NGRAPH * DIM, stream);
    hipMemsetAsync(cnt, 0, sizeof(float) * NGRAPH, stream);
    pool_sum<<<(ND + 255) / 256, 256, 0, stream>>>(hin, gid, hg, NNODES);
    pool_cnt<<<(NNODES + 255) / 256, 256, 0, stream>>>(gid, cnt, NNODES);
    pool_div<<<(NGRAPH * DIM + 255) / 256, 256, 0, stream>>>(hg, cnt);

    classifier_wmma<<<4 * 7, 32, 0, stream>>>(hg, Wc, bc, out);
}